// NonLinearConv2d_27144193310984
// MI455X (gfx1250) — compile-verified
//
#include <hip/hip_runtime.h>
#include <math.h>

// ---------------------------------------------------------------------------
// Problem constants (from reference)
// ---------------------------------------------------------------------------
#define N_BATCH   32
#define C_IN      3
#define HW        32
#define L_SPATIAL (HW * HW)          // 1024
#define C_OUT     64
#define K_TAPS    27                 // 3*3*3
#define NPOS      (N_BATCH * L_SPATIAL)  // 32768 positions per channel
#define OUT_ELEMS (N_BATCH * C_OUT * L_SPATIAL)

#define INV_DENOM 13.333333333333334f   // 1 / (2*1.5*0.025)
#define VD_SHIFT  1.3333333333333333f   // 0.1 * INV_DENOM
#define OUT_SCALE 5.625e-5f             // ALPHA * R_TIA
#define BN_EPS    1e-5f

typedef float v2f __attribute__((ext_vector_type(2)));
typedef float v8f __attribute__((ext_vector_type(8)));

__device__ __forceinline__ float softplus_f(float z) {
    // numerically stable: max(z,0) + log1p(exp(-|z|))
    return fmaxf(z, 0.0f) + log1pf(expf(-fabsf(z)));
}

// ---------------------------------------------------------------------------
// Kernel 1: nonlinear "conv".  Block = one (n, oh) row, 8 waves.
// Wave w handles channels [w*8, w*8+8); lane = ow (coalesced loads/stores).
// Patch values (27) loaded once per lane, reused across 8 channels.
// Theta is wave-uniform -> scalar-path reads from LDS.
// ---------------------------------------------------------------------------
__global__ __launch_bounds__(256) void nlconv_kernel(
    const float* __restrict__ x,      // (32,3,32,32)
    const float* __restrict__ theta,  // (27,64)
    float* __restrict__ out)          // (32,64,32,32)
{
    __shared__ float sTheta[K_TAPS * C_OUT];   // 6912 B
    for (int i = threadIdx.x; i < K_TAPS * C_OUT; i += 256)
        sTheta[i] = theta[i];
    __syncthreads();

    const int n    = blockIdx.x >> 5;
    const int oh   = blockIdx.x & 31;
    const int wave = threadIdx.x >> 5;
    const int lane = threadIdx.x & 31;

    const float* xb = x + n * (C_IN * L_SPATIAL);

    // gather 3x3x3 patch for output position (oh, lane), zero-padded
    float v[K_TAPS];
#pragma unroll
    for (int ci = 0; ci < 3; ++ci) {
#pragma unroll
        for (int ki = 0; ki < 3; ++ki) {
#pragma unroll
            for (int kj = 0; kj < 3; ++kj) {
                const int k  = (ci * 3 + ki) * 3 + kj;  // matches unfold order
                const int ih = oh + ki - 1;
                const int iw = lane + kj - 1;
                const bool ok = (ih >= 0) & (ih < HW) & (iw >= 0) & (iw < HW);
                v[k] = ok ? xb[ci * L_SPATIAL + ih * HW + iw] : 0.0f;
            }
        }
    }

    // 8 channels per wave, channel index is wave-uniform
    for (int cc = 0; cc < 8; ++cc) {
        const int c = wave * 8 + cc;
        float acc = 0.0f;
#pragma unroll
        for (int k = 0; k < K_TAPS; ++k) {
            const float th = sTheta[k * C_OUT + c];
            const float z1 = (v[k] - th) * INV_DENOM;
            const float z2 = z1 - VD_SHIFT;
            const float s1 = softplus_f(z1);
            const float s2 = softplus_f(z2);
            acc += s1 * s1 - s2 * s2;
        }
        out[((n * C_OUT + c) << 10) + (oh << 5) + lane] = acc * OUT_SCALE;
    }
}

// ---------------------------------------------------------------------------
// Kernel 2: per-channel sum / sumsq partials via V_WMMA_F32_16X16X4_F32.
// A-matrix rows = 16 channels, K = 4 consecutive positions, B = ones ->
// D[m][n] accumulates channel partial sums on the matrix pipe (co-executes
// with VALU).  Column N=0 (lanes 0 and 16) holds the per-channel results.
// Grid: 256 blocks = 4 channel-groups x 64 position-slices; 8 waves/block,
// each wave reduces 64 consecutive positions (16 WMMA steps x 2 matrices).
// Fixed-order -> deterministic.
// ---------------------------------------------------------------------------
__global__ __launch_bounds__(256) void stats_kernel(
    const float* __restrict__ vout,   // (32,64,32,32)
    float* __restrict__ psum,         // (64 slices, 64 channels)
    float* __restrict__ psumsq)       // (64 slices, 64 channels)
{
    const int cg   = blockIdx.x & 3;        // channel group (16 channels)
    const int ps   = blockIdx.x >> 2;       // position slice of this block
    const int wave = threadIdx.x >> 5;
    const int lane = threadIdx.x & 31;

    const int c0 = cg * 16;
    const int m  = lane & 15;               // channel row in A
    const int kb = (lane >> 4) * 2;         // k-slot base (0 or 2)
    const int c  = c0 + m;

    const int sbase = (ps * 8 + wave) * 64; // 64 consecutive positions

    v8f accS = {};                          // per-channel running sums
    v8f accQ = {};                          // per-channel running sum-of-squares
    v2f ones; ones.x = 1.0f; ones.y = 1.0f; // B matrix = all ones (4x16)

    for (int it = 0; it < 16; ++it) {
        const int p0 = sbase + it * 4 + kb; // position of this lane's k-pair
        const int n  = p0 >> 10;
        const int l  = p0 & 1023;           // l and l+1 stay in the same image
        const int addr = ((n * C_OUT + c) << 10) + l;

        v2f a;                              // A[m][kb], A[m][kb+1]
        a.x = vout[addr];
        a.y = vout[addr + 1];

        // D = A x ones + C : row sums accumulate into accS
        accS = __builtin_amdgcn_wmma_f32_16x16x4_f32(
            false, a, false, ones, (short)0, accS, false, false);

        v2f a2; a2.x = a.x * a.x; a2.y = a.y * a.y;
        accQ = __builtin_amdgcn_wmma_f32_16x16x4_f32(
            false, a2, false, ones, (short)0, accQ, false, false);
    }

    // Column N=0 lives in lane 0 (M=0..7 in vgpr 0..7) and lane 16 (M=8..15)
    __shared__ float sS[8][16];
    __shared__ float sQ[8][16];
    if (lane == 0) {
#pragma unroll
        for (int i = 0; i < 8; ++i) { sS[wave][i] = accS[i]; sQ[wave][i] = accQ[i]; }
    }
    if (lane == 16) {
#pragma unroll
        for (int i = 0; i < 8; ++i) { sS[wave][8 + i] = accS[i]; sQ[wave][8 + i] = accQ[i]; }
    }
    __syncthreads();

    if (threadIdx.x < 16) {
        float s = 0.0f, q = 0.0f;
        for (int wv = 0; wv < 8; ++wv) { s += sS[wv][threadIdx.x]; q += sQ[wv][threadIdx.x]; }
        psum  [ps * C_OUT + c0 + threadIdx.x] = s;
        psumsq[ps * C_OUT + c0 + threadIdx.x] = q;
    }
}

// ---------------------------------------------------------------------------
// Kernel 3: fold 64 slice-partials per channel (fixed order) -> scale/shift
// ---------------------------------------------------------------------------
__global__ __launch_bounds__(64) void finalize_kernel(
    const float* __restrict__ psum,
    const float* __restrict__ psumsq,
    const float* __restrict__ gamma,
    const float* __restrict__ beta,
    float* __restrict__ scale,
    float* __restrict__ shift)
{
    const int c = threadIdx.x;
    float s = 0.0f, q = 0.0f;
    for (int ps = 0; ps < 64; ++ps) {
        s += psum  [ps * C_OUT + c];
        q += psumsq[ps * C_OUT + c];
    }
    const float inv_n = 1.0f / (float)NPOS;
    const float mean  = s * inv_n;
    const float var   = q * inv_n - mean * mean;   // biased (training mode)
    const float sc    = gamma[c] * rsqrtf(var + BN_EPS);
    scale[c] = sc;
    shift[c] = beta[c] - mean * sc;
}

// ---------------------------------------------------------------------------
// Kernel 4: in-place BN apply, fully coalesced
// ---------------------------------------------------------------------------
__global__ __launch_bounds__(256) void bn_apply_kernel(
    float* __restrict__ out,
    const float* __restrict__ scale,
    const float* __restrict__ shift)
{
    const int idx = blockIdx.x * 256 + threadIdx.x;
    if (idx >= OUT_ELEMS) return;
    const int c = (idx >> 10) & (C_OUT - 1);
    out[idx] = fmaf(out[idx], scale[c], shift[c]);
}

// ---------------------------------------------------------------------------
extern "C" void kernel_launch(void* const* d_in, const int* in_sizes, int n_in,
                              void* d_out, int out_size, void* d_ws, size_t ws_size,
                              hipStream_t stream)
{
    const float* x     = (const float*)d_in[0];
    const float* theta = (const float*)d_in[1];
    const float* gamma = (const float*)d_in[2];
    const float* beta  = (const float*)d_in[3];
    float* out = (float*)d_out;

    // workspace layout (floats): 64x64 psum | 64x64 psumsq | 64 scale | 64 shift
    float* ws     = (float*)d_ws;
    float* psum   = ws;
    float* psumsq = ws + 64 * 64;
    float* scale  = ws + 2 * 64 * 64;
    float* shift  = ws + 2 * 64 * 64 + 64;

    nlconv_kernel  <<<N_BATCH * HW, 256, 0, stream>>>(x, theta, out);
    stats_kernel   <<<256,          256, 0, stream>>>(out, psum, psumsq);
    finalize_kernel<<<1,             64, 0, stream>>>(psum, psumsq, gamma, beta, scale, shift);
    bn_apply_kernel<<<(OUT_ELEMS + 255) / 256, 256, 0, stream>>>(out, scale, shift);
}